// ObjectDetectionModel_15187004359198
// MI455X (gfx1250) — compile-verified
//
#include <hip/hip_runtime.h>
#include <math.h>

#define B_   8
#define C_   256
#define H_   192
#define W_   192
#define K_   32
#define HW_  (H_*W_)
#define NPIX (B_*HW_)      // 294912
#define NROW 248           // 8*31 rows of vec
#define C2   128

typedef __attribute__((ext_vector_type(2))) float v2f;
typedef __attribute__((ext_vector_type(8))) float v8f;

// workspace layout
#define WS_TAB_OFF (512*1024)   // int/float [B_*K_*C_] = 65536 entries
#define WS_H_OFF   (1024*1024)  // float [NROW*C2]

// ---------------- kernel 1: softmax over K + argmax -> seg ids ----------------
__global__ __launch_bounds__(256)
void k_softmax_argmax(const float* __restrict__ logits,
                      float* __restrict__ out_probs,
                      unsigned char* __restrict__ seg) {
  int p = blockIdx.x * blockDim.x + threadIdx.x;
  if (p >= NPIX) return;
  int b  = p / HW_;
  int hw = p - b * HW_;
  long base = (long)b * K_ * HW_ + hw;
  float x[K_];
#pragma unroll
  for (int k = 0; k < K_; ++k) x[k] = logits[base + (long)k * HW_];
  float m = x[0]; int am = 0;
#pragma unroll
  for (int k = 1; k < K_; ++k) { if (x[k] > m) { m = x[k]; am = k; } }
  float s = 0.f;
#pragma unroll
  for (int k = 0; k < K_; ++k) { float e = expf(x[k] - m); x[k] = e; s += e; }
  float inv = 1.0f / s;
#pragma unroll
  for (int k = 0; k < K_; ++k) out_probs[base + (long)k * HW_] = x[k] * inv;
  seg[p] = (unsigned char)am;
}

// ---------------- kernel 2: zero the segment-max table ----------------
__global__ void k_zero_table(int* __restrict__ tab) {
  int i = blockIdx.x * blockDim.x + threadIdx.x;
  if (i < B_ * K_ * C_) tab[i] = 0;
}

// ---------------- kernel 3: segment max via LDS pre-reduction ----------------
#define TILE_PIX 1024
#define TILES_PER_B (HW_ / TILE_PIX)   // 36

__global__ __launch_bounds__(256)
void k_segmax(const float* __restrict__ enc,
              const unsigned char* __restrict__ seg,
              int* __restrict__ gtab) {
  __shared__ int lds[K_ * 257];        // padded stride 257 -> bank (s+c)%64
  int tid = threadIdx.x;
  for (int e = tid; e < K_ * 257; e += 256) lds[e] = 0;

  int bid = blockIdx.x;
  int b = bid / TILES_PER_B;
  int t = bid - b * TILES_PER_B;
  int pbase = t * TILE_PIX + tid * 4;  // pixel offset within batch

  const unsigned char* sp = seg + b * HW_ + pbase;
  int s0 = sp[0], s1 = sp[1], s2 = sp[2], s3 = sp[3];
  __syncthreads();

  const float* ep = enc + (long)b * C_ * HW_ + pbase;
  for (int c = 0; c < C_; ++c) {
    float4 v = *(const float4*)(ep + (long)c * HW_);   // coalesced 16B/lane
    atomicMax(&lds[s0 * 257 + c], __float_as_int(fmaxf(v.x, 0.f)));
    atomicMax(&lds[s1 * 257 + c], __float_as_int(fmaxf(v.y, 0.f)));
    atomicMax(&lds[s2 * 257 + c], __float_as_int(fmaxf(v.z, 0.f)));
    atomicMax(&lds[s3 * 257 + c], __float_as_int(fmaxf(v.w, 0.f)));
  }
  __syncthreads();

  for (int e = tid; e < K_ * C_; e += 256) {
    int s = e >> 8, c = e & 255;
    int v = lds[s * 257 + c];
    if (v != 0) atomicMax(&gtab[(b * K_ + s) * C_ + c], v);
  }
}

// ---------------- kernel 4: h = vec @ w1 + b1 via v_wmma_f32_16x16x4_f32 ----
// Padded rows (r >= 248) only affect discarded D rows -> clamp index, no mask.
__global__ __launch_bounds__(32)
void k_gemm1(const float* __restrict__ tabF,  // [B][32][256] relu'd seg max
             const float* __restrict__ w1,    // [256][128]
             const float* __restrict__ b1,    // [128]
             float* __restrict__ h) {         // [248][128]
  int lane = threadIdx.x;
  int mt = blockIdx.x >> 3;       // 16 M tiles
  int nt = blockIdx.x & 7;        // 8 N tiles
  int m0 = mt * 16, n0 = nt * 16;
  int half = lane >> 4;           // K-pair select
  int l = lane & 15;
  int rA = m0 + l;
  int rc = (rA < NROW) ? rA : 0;  // clamp: garbage only lands in dropped rows
  int bA = rc / 31, sA = rc - bA * 31;
  long aBase = (long)(bA * K_ + 1 + sA) * C_;   // skip segment 0
  int kb = half * 2;
  int col = l;

  v8f acc = {};
  for (int k0 = 0; k0 < C_; k0 += 4) {
    v2f a = *(const v2f*)(tabF + aBase + k0 + kb);   // 8B-aligned pair load
    v2f bm;
    bm.x = w1[(long)(k0 + kb)     * C2 + n0 + col];
    bm.y = w1[(long)(k0 + kb + 1) * C2 + n0 + col];
    acc = __builtin_amdgcn_wmma_f32_16x16x4_f32(false, a, false, bm,
                                                (short)0, acc, false, false);
  }
  float bias = b1[n0 + col];
#pragma unroll
  for (int i = 0; i < 8; ++i) {
    int r = m0 + i + half * 8;     // VGPR i: rows i (lanes 0-15), i+8 (16-31)
    if (r < NROW) h[r * C2 + n0 + col] = acc[i] + bias;
  }
}

// ---------------- kernel 5: out = sigmoid(h @ w2 + b2) via WMMA -------------
// Padded rows/cols feed only discarded D entries -> clamp indices, no masks.
__global__ __launch_bounds__(32)
void k_gemm2(const float* __restrict__ h,     // [248][128]
             const float* __restrict__ w2,    // [128][4]
             const float* __restrict__ b2,    // [4]
             float* __restrict__ out) {       // [248][4]
  int lane = threadIdx.x;
  int m0 = blockIdx.x * 16;        // 16 M tiles
  int half = lane >> 4;
  int l = lane & 15;
  int rA = m0 + l;
  int rc = (rA < NROW) ? rA : 0;
  int kb = half * 2;
  int col = l;
  int cc = (col < 4) ? col : 0;

  v8f acc = {};
  for (int k0 = 0; k0 < C2; k0 += 4) {
    v2f a = *(const v2f*)(h + rc * C2 + k0 + kb);    // 8B-aligned pair load
    v2f bm;
    bm.x = w2[(k0 + kb)     * 4 + cc];
    bm.y = w2[(k0 + kb + 1) * 4 + cc];
    acc = __builtin_amdgcn_wmma_f32_16x16x4_f32(false, a, false, bm,
                                                (short)0, acc, false, false);
  }
  float bias = b2[cc];
#pragma unroll
  for (int i = 0; i < 8; ++i) {
    int r = m0 + i + half * 8;
    if (r < NROW && col < 4) {
      float x = acc[i] + bias;
      out[r * 4 + col] = 1.0f / (1.0f + expf(-x));
    }
  }
}

extern "C" void kernel_launch(void* const* d_in, const int* in_sizes, int n_in,
                              void* d_out, int out_size, void* d_ws, size_t ws_size,
                              hipStream_t stream) {
  const float* enc    = (const float*)d_in[0];
  const float* logits = (const float*)d_in[1];
  const float* w1     = (const float*)d_in[2];
  const float* b1     = (const float*)d_in[3];
  const float* w2     = (const float*)d_in[4];
  const float* b2     = (const float*)d_in[5];

  float* outB = (float*)d_out;            // out_bboxes: 248*4 = 992 floats
  float* outP = outB + NROW * 4;          // encoded_classes: 8*32*192*192

  unsigned char* seg = (unsigned char*)d_ws;                 // [NPIX] bytes
  int*   tab  = (int*)((char*)d_ws + WS_TAB_OFF);            // [8*32*256]
  float* tabF = (float*)tab;                                 // same bits
  float* hbuf = (float*)((char*)d_ws + WS_H_OFF);            // [248*128]

  k_softmax_argmax<<<NPIX / 256, 256, 0, stream>>>(logits, outP, seg);
  k_zero_table<<<(B_ * K_ * C_) / 256, 256, 0, stream>>>(tab);
  k_segmax<<<B_ * TILES_PER_B, 256, 0, stream>>>(enc, seg, tab);
  k_gemm1<<<128, 32, 0, stream>>>(tabF, w1, b1, hbuf);
  k_gemm2<<<16, 32, 0, stream>>>(hbuf, w2, b2, outB);
}